// MaxpoolReadoutLayer_81243601371198
// MI455X (gfx1250) — compile-verified
//
#include <hip/hip_runtime.h>
#include <hip/hip_bf16.h>
#include <stdint.h>

// Problem constants (from reference): B=64, N=4096, D=256.
#define B_  64
#define N_  4096
#define D_  256
#define SPLIT_ 16                  // blocks per batch along N
#define ROWS_PER_BLOCK_ (N_ / SPLIT_)          // 256
#define STAGE_ROWS_ 16                          // rows per LDS stage
#define STAGE_FLOATS_ (STAGE_ROWS_ * D_)        // 4096 floats = 16 KB
#define ISSUES_PER_STAGE_ 4                     // 256 thr * 16B * 4 = 16 KB
#define STAGES_MAX_ (ROWS_PER_BLOCK_ / STAGE_ROWS_)  // 16
#define NEGSENT_ (-3.0e38f)

// ROCm 7.2 clang-22 signature of the async-copy builtin takes v4i pointers:
//   (v4i addrspace(1)* gsrc, v4i addrspace(3)* ldst, imm offset, imm cpol)
typedef int v4i_ __attribute__((ext_vector_type(4)));
typedef __attribute__((address_space(1))) v4i_ gv4i_t;
typedef __attribute__((address_space(3))) v4i_ lv4i_t;

// ---- CDNA5 async global->LDS copy (ASYNCcnt-tracked, per-lane 16B) ----
__device__ __forceinline__ void async_copy_b128(const float* gsrc, float* ldst) {
#if __has_builtin(__builtin_amdgcn_global_load_async_to_lds_b128)
  __builtin_amdgcn_global_load_async_to_lds_b128((gv4i_t*)gsrc, (lv4i_t*)ldst,
                                                 /*offset=*/0, /*cpol=*/0);
#else
  unsigned lofs = (unsigned)(unsigned long long)ldst;  // low 32 bits = LDS offset
  asm volatile("global_load_async_to_lds_b128 %0, %1, off"
               :: "v"(lofs), "v"(gsrc) : "memory");
#endif
}

template <int CNT>
__device__ __forceinline__ void wait_asynccnt() {
#if __has_builtin(__builtin_amdgcn_s_wait_asynccnt)
  __builtin_amdgcn_s_wait_asynccnt(CNT);
#else
  asm volatile("s_wait_asynccnt %0" :: "n"(CNT) : "memory");
#endif
}

// Order-preserving f32 -> u32 map for integer max.
__device__ __forceinline__ unsigned map_f32(float v) {
  unsigned u = __float_as_uint(v);
  return (u & 0x80000000u) ? ~u : (u | 0x80000000u);
}
__device__ __forceinline__ float unmap_f32(unsigned m) {
  unsigned u = (m & 0x80000000u) ? (m & 0x7fffffffu) : ~m;
  return __uint_as_float(u);
}
// key = [ mapped value | 0xFFFFFFFF - n ]  => max key = max value, smallest n on ties.
__device__ __forceinline__ unsigned long long pack_key(float v, int n) {
  return ((unsigned long long)map_f32(v) << 32) |
         (unsigned long long)(0xFFFFFFFFu - (unsigned)n);
}

__global__ __launch_bounds__(256) void mr_init_kernel(unsigned long long* keys) {
  int i = blockIdx.x * 256 + threadIdx.x;
  if (i < B_ * D_) keys[i] = 0ull;
}

__global__ __launch_bounds__(256) void mr_main_kernel(const float* __restrict__ hidden,
                                                      const int* __restrict__ child_counts,
                                                      unsigned long long* __restrict__ keys) {
  __shared__ alignas(16) float sbuf[2 * STAGE_FLOATS_];      // 32 KB double buffer
  __shared__ unsigned long long keybuf[256 * 4];             // 8 KB

  const int tid   = threadIdx.x;
  const int split = blockIdx.x;
  const int b     = blockIdx.y;

  const int cc  = child_counts[b];
  const int eff = (cc == 0) ? 1 : cc;                        // valid rows
  const int chunk = split * ROWS_PER_BLOCK_;

  int rows = eff - chunk;
  if (rows <= 0) return;                                     // whole chunk masked (uniform exit)
  if (rows > ROWS_PER_BLOCK_) rows = ROWS_PER_BLOCK_;
  const int stages = (rows + STAGE_ROWS_ - 1) >> 4;          // ceil(rows/16)

  const int q = tid & 63;         // d-quad id
  const int r0 = tid >> 6;        // row phase 0..3
  const int dbase = q * 4;

  float m0 = NEGSENT_, m1 = NEGSENT_, m2 = NEGSENT_, m3 = NEGSENT_;
  int   i0 = 0, i1 = 0, i2 = 0, i3 = 0;

  const float* gchunk = hidden + ((size_t)b * N_ + chunk) * D_;

  auto issue_stage = [&](int s) {
    const float* gb = gchunk + (size_t)s * STAGE_FLOATS_;
    float* lb = &sbuf[(s & 1) * STAGE_FLOATS_];
#pragma unroll
    for (int i = 0; i < ISSUES_PER_STAGE_; ++i) {
      const int fo = i * 1024 + tid * 4;                     // 4 KB per issue across block
      async_copy_b128(gb + fo, lb + fo);
    }
  };

  auto compute_stage = [&](int s) {
    const float* bp = &sbuf[(s & 1) * STAGE_FLOATS_];
    const int nb = chunk + s * STAGE_ROWS_;
#pragma unroll
    for (int j = 0; j < 4; ++j) {
      const int r = r0 + 4 * j;
      const int n = nb + r;
      if (n < eff) {                                         // wave-uniform predicate
        const float4 v = *(const float4*)(bp + r * D_ + dbase);
        if (v.x > m0) { m0 = v.x; i0 = n; }
        if (v.y > m1) { m1 = v.y; i1 = n; }
        if (v.z > m2) { m2 = v.z; i2 = n; }
        if (v.w > m3) { m3 = v.w; i3 = n; }
      }
    }
  };

  // Software pipeline: async-fill next stage while reducing current one.
  issue_stage(0);
  int s = 0;
  for (; s + 1 < stages; ++s) {
    issue_stage(s + 1);
    wait_asynccnt<ISSUES_PER_STAGE_>();   // current stage landed (in-order completion)
    __syncthreads();
    compute_stage(s);
    __syncthreads();
  }
  wait_asynccnt<0>();
  __syncthreads();
  compute_stage(s);

  // Intra-block reduce across the 4 row phases, then one atomic per d.
  keybuf[tid * 4 + 0] = (m0 > NEGSENT_ * 0.5f) ? pack_key(m0, i0) : 0ull;
  keybuf[tid * 4 + 1] = (m1 > NEGSENT_ * 0.5f) ? pack_key(m1, i1) : 0ull;
  keybuf[tid * 4 + 2] = (m2 > NEGSENT_ * 0.5f) ? pack_key(m2, i2) : 0ull;
  keybuf[tid * 4 + 3] = (m3 > NEGSENT_ * 0.5f) ? pack_key(m3, i3) : 0ull;
  __syncthreads();
  if (tid < 64) {
#pragma unroll
    for (int k = 0; k < 4; ++k) {
      unsigned long long best = keybuf[tid * 4 + k];
      unsigned long long c1 = keybuf[(tid + 64) * 4 + k];  if (c1 > best) best = c1;
      unsigned long long c2 = keybuf[(tid + 128) * 4 + k]; if (c2 > best) best = c2;
      unsigned long long c3 = keybuf[(tid + 192) * 4 + k]; if (c3 > best) best = c3;
      if (best) atomicMax(&keys[b * D_ + tid * 4 + k], best);
    }
  }
}

__global__ __launch_bounds__(256) void mr_decode_kernel(const unsigned long long* __restrict__ keys,
                                                        float* __restrict__ vals,
                                                        int* __restrict__ idx_out) {
  int i = blockIdx.x * 256 + threadIdx.x;
  if (i >= B_ * D_) return;
  const unsigned long long key = keys[i];
  const unsigned mm = (unsigned)(key >> 32);
  const unsigned n  = 0xFFFFFFFFu - (unsigned)(key & 0xFFFFFFFFu);
  const int d = i & (D_ - 1);
  vals[i]    = unmap_f32(mm);
  idx_out[i] = (int)n * D_ + d;    // flat index into the (N, D) plane
}

extern "C" void kernel_launch(void* const* d_in, const int* in_sizes, int n_in,
                              void* d_out, int out_size, void* d_ws, size_t ws_size,
                              hipStream_t stream) {
  (void)in_sizes; (void)n_in; (void)out_size; (void)ws_size;
  const float* hidden       = (const float*)d_in[0];
  const int*   child_counts = (const int*)d_in[1];

  float* vals  = (float*)d_out;                 // [B*D] f32
  int*   idxo  = (int*)d_out + B_ * D_;         // [B*D] i32 (flat n*D+d)
  unsigned long long* keys = (unsigned long long*)d_ws;   // 128 KB scratch

  mr_init_kernel<<<(B_ * D_ + 255) / 256, 256, 0, stream>>>(keys);
  mr_main_kernel<<<dim3(SPLIT_, B_), 256, 0, stream>>>(hidden, child_counts, keys);
  mr_decode_kernel<<<(B_ * D_ + 255) / 256, 256, 0, stream>>>(keys, vals, idxo);
}